// Nbr_Center_SpectralNetLoss_29377576305179
// MI455X (gfx1250) — compile-verified
//
#include <hip/hip_runtime.h>
#include <hip/hip_bf16.h>
#include <math.h>

// Shapes (fixed by the reference):
//   M=1024, D=128, BK=8, BCK=1, N_NBR=8192, N_CORE=8192
// Inputs: 0 W_nbr[1024,8192] f32, 1 W_core[8192,8192] f32, 2 Y[1024,128] f32,
//         3 Y_nbr[8192,128] f32, 4 Y_core[8192,128] f32, 5 LE[8192,128] f32,
//         6 lamda f32, 7 bk i32, 8 bck i32.  Output: 1 f32 scalar.

typedef float v2f __attribute__((ext_vector_type(2)));
typedef float v8f __attribute__((ext_vector_type(8)));

#define D_DIM      128
#define ROWS       8192
#define ROW_STRIDE 8192   // W matrices' second dim
#define BLOCKS_PER_TERM 64
#define NBLOCKS    (3 * BLOCKS_PER_TERM)

// One block = 8 waves; each wave computes the squared L2 distances of 16
// row-pairs via 32 chained V_WMMA_F32_16X16X4_F32 Gram-matrix accumulations
// (diagonal of diff * diff^T = squared norms), then weights + reduces.
__global__ __launch_bounds__(256) void spectral_dist_kernel(
    const float* __restrict__ W_nbr,
    const float* __restrict__ W_core,
    const float* __restrict__ Y,
    const float* __restrict__ Y_nbr,
    const float* __restrict__ Y_core,
    const float* __restrict__ LE,
    float* __restrict__ partials)
{
    const int lane = threadIdx.x & 31;
    const int wid  = threadIdx.x >> 5;
    const int term = blockIdx.x >> 6;        // 0,1,2  (uniform per block)
    const int blk  = blockIdx.x & 63;
    const int R0   = blk * 128 + wid * 16;   // first row of this wave's 16-row tile
    const int row  = R0 + (lane & 15);       // row this lane streams
    const int colShift = (lane >> 4) << 1;   // 0 for lanes 0-15, 2 for lanes 16-31

    // Pick the two vector sources for this term (uniform branch -> EXEC stays full).
    const float* Xrow;
    const float* Zrow;
    if (term == 0) {
        Xrow = Y      + (size_t)(row >> 3) * D_DIM;   // Y row reused by 8 neighbors
        Zrow = Y_nbr  + (size_t)row * D_DIM;
    } else if (term == 1) {
        Xrow = Y_nbr  + (size_t)row * D_DIM;
        Zrow = Y_core + (size_t)row * D_DIM;
    } else {
        Xrow = Y_core + (size_t)row * D_DIM;
        Zrow = LE     + (size_t)row * D_DIM;
    }

    // Gram accumulation: A(16x4) = diff chunk, B(4x16) = diff^T chunk.
    // f32 WMMA A/B lane layouts are symmetric (lane L: M/N = L%16,
    // K = colShift + {0,1}), so the SAME float2 register feeds A and B.
    v8f c = {0.f, 0.f, 0.f, 0.f, 0.f, 0.f, 0.f, 0.f};
    #pragma unroll
    for (int ch = 0; ch < D_DIM / 4; ++ch) {
        const int d0 = ch * 4 + colShift;
        v2f x = *(const v2f*)(Xrow + d0);
        v2f z = *(const v2f*)(Zrow + d0);
        v2f a = x - z;
        // D = A*B + C : diagonal accumulates sum_k diff[m][k]^2
        c = __builtin_amdgcn_wmma_f32_16x16x4_f32(
                /*neg_a=*/false, a, /*neg_b=*/false, a,
                /*c_mod=*/(short)0, c, /*reuse_a=*/false, /*reuse_b=*/false);
    }

    // Diagonal element m lives at (VGPR m%8, lane m<8 ? m : m+16).
    const bool active = (lane < 8) || (lane >= 24);
    int sel = (lane < 8) ? lane : ((lane >= 24) ? (lane - 24) : 0);
    float d2 = c[0];
    #pragma unroll
    for (int i = 1; i < 8; ++i)
        if (sel == i) d2 = c[i];

    float val = 0.f;
    if (active) {
        const int m = (lane < 8) ? lane : (lane - 16);  // diag index 0..15
        const int r = R0 + m;                            // global row-pair id
        if (term == 0)       val = W_nbr[(size_t)(r >> 3) * ROW_STRIDE + r] * d2;
        else if (term == 1)  val = W_core[(size_t)r * (ROW_STRIDE + 1)] * d2;
        else                 val = sqrtf(d2);
    }

    // Fixed-order wave reduction (deterministic), then fixed-order block combine.
    #pragma unroll
    for (int off = 16; off > 0; off >>= 1)
        val += __shfl_xor(val, off, 32);

    __shared__ float wsum[8];
    if (lane == 0) wsum[wid] = val;
    __syncthreads();
    if (threadIdx.x == 0) {
        float t = 0.f;
        #pragma unroll
        for (int i = 0; i < 8; ++i) t += wsum[i];
        partials[blockIdx.x] = t;
    }
}

// Deterministic final combine of the 192 block partials.
__global__ void spectral_reduce_kernel(const float* __restrict__ partials,
                                       const float* __restrict__ lamda,
                                       float* __restrict__ out)
{
    if (threadIdx.x == 0 && blockIdx.x == 0) {
        float t1 = 0.f, t2 = 0.f, t3 = 0.f;
        for (int i = 0; i < BLOCKS_PER_TERM; ++i)                          t1 += partials[i];
        for (int i = BLOCKS_PER_TERM; i < 2 * BLOCKS_PER_TERM; ++i)        t2 += partials[i];
        for (int i = 2 * BLOCKS_PER_TERM; i < 3 * BLOCKS_PER_TERM; ++i)    t3 += partials[i];
        // t1/l1 + t2/l2 + lamda * t3 / n_core  with l1 = l2 = n_core = 8192
        out[0] = (t1 + t2 + lamda[0] * t3) * (1.0f / 8192.0f);
    }
}

extern "C" void kernel_launch(void* const* d_in, const int* in_sizes, int n_in,
                              void* d_out, int out_size, void* d_ws, size_t ws_size,
                              hipStream_t stream)
{
    const float* W_nbr  = (const float*)d_in[0];
    const float* W_core = (const float*)d_in[1];
    const float* Y      = (const float*)d_in[2];
    const float* Y_nbr  = (const float*)d_in[3];
    const float* Y_core = (const float*)d_in[4];
    const float* LE     = (const float*)d_in[5];
    const float* lamda  = (const float*)d_in[6];

    float* partials = (float*)d_ws;   // NBLOCKS floats of scratch

    spectral_dist_kernel<<<NBLOCKS, 256, 0, stream>>>(
        W_nbr, W_core, Y, Y_nbr, Y_core, LE, partials);
    spectral_reduce_kernel<<<1, 32, 0, stream>>>(partials, lamda, (float*)d_out);
}